// BinaryPooling2d_88510686036631
// MI455X (gfx1250) — compile-verified
//
#include <hip/hip_runtime.h>
#include <stdint.h>

typedef float v2f __attribute__((ext_vector_type(2)));
typedef float v8f __attribute__((ext_vector_type(8)));

#define HO   126
#define NPIX (126 * 126)

__global__ __launch_bounds__(256)
void binary_pool_kernel(const float* __restrict__ x, float* __restrict__ out) {
    __shared__ float smem[128 * 128];   // one (b,c) slice, 64 KB
    __shared__ float red[8];            // per-wave partials

    const int tid = threadIdx.x;
    const unsigned long long base =
        (unsigned long long)(uintptr_t)(x + (size_t)blockIdx.x * (128 * 128));

    // ---- Stage the 64KB slice into LDS via CDNA5 async-to-LDS DMA ----
    // GVS mode: mem_addr = SGPR64 base + VGPR32 offset; VDST vgpr = LDS byte addr.
    // Generic LDS pointers: low 32 bits are the LDS offset (aperture layout).
    const unsigned ldsBase = (unsigned)(uintptr_t)&smem[0];
    for (int t = tid; t < 4096; t += 256) {           // 4096 * 16B = 64KB
        unsigned off = (unsigned)t * 16u;
        asm volatile("global_load_async_to_lds_b128 %0, %1, %2"
                     :: "v"(ldsBase + off), "v"(off), "s"(base)
                     : "memory");
    }
    asm volatile("s_wait_asynccnt 0x0" ::: "memory");
    __syncthreads();

    // ---- Fused 3x3 stencil from LDS, per-thread accumulation ----
    float acc = 0.0f;
    for (int p = tid; p < NPIX; p += 256) {
        const int i = p / HO;
        const int j = p - i * HO;
        const float* r = &smem[i * 128 + j];
        const float v0 = r[0],   v1 = r[1],   v2 = r[2];
        const float v3 = r[128], v4 = r[129], v5 = r[130];
        const float v6 = r[256], v7 = r[257], v8 = r[258];

        const float s  = ((v0 + v1) + (v2 + v3)) + ((v5 + v6) + (v7 + v8)) + v4;
        const float ss = fmaf(v0, v0, fmaf(v1, v1, fmaf(v2, v2, fmaf(v3, v3,
                         fmaf(v4, v4, fmaf(v5, v5, fmaf(v6, v6, fmaf(v7, v7,
                         v8 * v8))))))));
        const float mx = fmaxf(fmaxf(fmaxf(fmaxf(v0, v1), fmaxf(v2, v3)),
                               fmaxf(fmaxf(v5, v6), fmaxf(v7, v8))), v4);
        const float c  = v4;
        const float sad = (fabsf(v0 - c) + fabsf(v1 - c)) +
                          (fabsf(v2 - c) + fabsf(v3 - c)) +
                          (fabsf(v5 - c) + fabsf(v6 - c)) +
                          (fabsf(v7 - c) + fabsf(v8 - c));
        const float thr = sad * (1.0f / 9.0f);

        float cnt = 0.0f;
        cnt += (v0 - c >= thr) ? 1.0f : 0.0f;
        cnt += (v1 - c >= thr) ? 1.0f : 0.0f;
        cnt += (v2 - c >= thr) ? 1.0f : 0.0f;
        cnt += (v3 - c >= thr) ? 1.0f : 0.0f;
        cnt += (0.0f    >= thr) ? 1.0f : 0.0f;   // center diff == 0
        cnt += (v5 - c >= thr) ? 1.0f : 0.0f;
        cnt += (v6 - c >= thr) ? 1.0f : 0.0f;
        cnt += (v7 - c >= thr) ? 1.0f : 0.0f;
        cnt += (v8 - c >= thr) ? 1.0f : 0.0f;

        const float mean = s * (1.0f / 9.0f);
        float var = fmaf(-mean, mean, ss * (1.0f / 9.0f));
        var = fmaxf(var, 0.0f);
        const float sd = __builtin_sqrtf(var);
        const float nf = (cnt - mean) * (1.0f / 255.0f);
        // nf*sd + (1-nf)*mx == mx + nf*(sd - mx)
        acc += fmaf(nf, sd - mx, mx);
    }

    // ---- Wave reduction via V_WMMA_F32_16X16X4_F32 with B = ones ----
    // A layout (16x4 f32): lanes 0-15 VGPR0 = A[m,0], VGPR1 = A[m,1];
    //                      lanes 16-31 VGPR0 = A[m,2], VGPR1 = A[m,3].
    // With A-VGPR0 = acc, A-VGPR1 = 0, B = 1: D[m,n] = acc[m] + acc[m+16].
    v2f a; a[0] = acc;  a[1] = 0.0f;
    v2f b; b[0] = 1.0f; b[1] = 1.0f;
    v8f cz = {};
    v8f d = __builtin_amdgcn_wmma_f32_16x16x4_f32(
        /*neg_a=*/false, a, /*neg_b=*/false, b,
        /*c_mod=*/(short)0, cz, /*reuse_a=*/false, /*reuse_b=*/false);

    float ws = 0.0f;
#pragma unroll
    for (int k = 0; k < 8; ++k) ws += d[k];     // lanes 0-15: rows 0-7; lanes 16-31: rows 8-15
    ws += __shfl_xor(ws, 16, 32);               // full 32-lane sum in every lane

    const int wave = tid >> 5;
    const int lane = tid & 31;
    if (lane == 0) red[wave] = ws;
    __syncthreads();

    if (tid == 0) {
        float tot = 0.0f;
#pragma unroll
        for (int k = 0; k < 8; ++k) tot += red[k];
        out[blockIdx.x] = tot * (1.0f / (float)NPIX);
    }
}

extern "C" void kernel_launch(void* const* d_in, const int* in_sizes, int n_in,
                              void* d_out, int out_size, void* d_ws, size_t ws_size,
                              hipStream_t stream) {
    const float* x = (const float*)d_in[0];
    float* out = (float*)d_out;
    // One workgroup per (b,c) slice: out_size == 16*64 == 1024.
    binary_pool_kernel<<<dim3((unsigned)out_size), dim3(256), 0, stream>>>(x, out);
}